// PrRoIPool_11630771438414
// MI455X (gfx1250) — compile-verified
//
#include <hip/hip_runtime.h>
#include <hip/hip_bf16.h>

typedef float v2f __attribute__((ext_vector_type(2)));
typedef float v8f __attribute__((ext_vector_type(8)));

#define POOLED 7
#define SCALE  0.0625f
#define Cc 256
#define Hh 100
#define Ww 100
#define WSMAX 104   // max w-support (<=100), padded to multiple of 4
#define HSMAX 104   // max h-support

// antiderivative of hat phi(u)=max(0,1-|u|), Phi(-inf)=0 (matches reference)
__device__ __forceinline__ float hat_cdf(float u) {
  float a = fminf(fmaxf(u + 1.0f, 0.0f), 1.0f);
  float b = fminf(fmaxf(1.0f - u, 0.0f), 1.0f);
  return (u <= 0.0f) ? (0.5f * a * a) : (1.0f - 0.5f * b * b);
}

__global__ __launch_bounds__(64)
void prroi_pool_wmma(const float* __restrict__ F,
                     const float* __restrict__ rois,
                     float* __restrict__ out) {
  // T[p][c][w]: h-reduced partial integrals (A-matrix for stage B)
  __shared__ float sT[POOLED][16][WSMAX];
  // wx^T[w][q], q padded to 16 (B-matrix for stage B)
  __shared__ float sWxT[WSMAX][16];
  // wy[p][h - hlo]
  __shared__ float sWy[POOLED][HSMAX];

  const int n    = blockIdx.y;         // roi
  const int c0   = blockIdx.x * 16;    // channel tile base
  const int tid  = threadIdx.x;        // 0..63 (2 waves)
  const int lane = tid & 31;
  const int wave = tid >> 5;

  const float* r = rois + n * 5;
  const int   bi  = (int)r[0];
  const float x1  = r[1] * SCALE, y1v = r[2] * SCALE;
  const float x2  = r[3] * SCALE, y2v = r[4] * SCALE;
  const float bw  = (x2 - x1) / (float)POOLED;
  const float bh  = (y2v - y1v) / (float)POOLED;
  const float area = bw * bh;
  const float invA = (area > 0.0f) ? (1.0f / fmaxf(area, 1e-12f)) : 0.0f;

  int wlo = max(0, (int)floorf(x1 - 1.0f));
  int whi = min(Ww - 1, (int)ceilf(x2 + 1.0f));
  int WS  = whi - wlo + 1; if (WS < 0) WS = 0;
  const int WS4 = min(WSMAX, (WS + 3) & ~3);

  int hlo = max(0, (int)floorf(y1v - 1.0f));
  int hhi = min(Hh - 1, (int)ceilf(y2v + 1.0f));

  // ---- weights into LDS ----
  for (int idx = tid; idx < WSMAX * 16; idx += 64) {
    int w = idx >> 4, q = idx & 15;
    float v = 0.0f;
    if (q < POOLED && w < WS) {
      float xloq = x1 + (float)q * bw;
      float xhiq = xloq + bw;
      float wg   = (float)(wlo + w);
      v = hat_cdf(xhiq - wg) - hat_cdf(xloq - wg);
    }
    sWxT[w][q] = v;
  }
  for (int idx = tid; idx < POOLED * HSMAX; idx += 64) {
    int p  = idx / HSMAX, hh = idx % HSMAX;
    int hg = hlo + hh;
    float v = 0.0f;
    if (hg <= hhi) {
      float ylop = y1v + (float)p * bh;
      float yhip = ylop + bh;
      v = hat_cdf(yhip - (float)hg) - hat_cdf(ylop - (float)hg);
    }
    sWy[p][hh] = v;
  }
  __syncthreads();

  // ---- Stage A (VALU): T[p][c][w] = sum_h wy[p][h] * F[bi, c0+c, h, wlo+w]
  // per-bin sparse h support (~bh+2 rows); coalesced w-contiguous F loads
  for (int ci = 0; ci < 8; ++ci) {
    int c = wave * 8 + ci;
    const float* Fc = F + (((size_t)bi * Cc + (size_t)(c0 + c)) * Hh) * Ww;
    for (int p = 0; p < POOLED; ++p) {
      float ylop = y1v + (float)p * bh;
      float yhip = ylop + bh;
      int h0 = max(hlo, (int)floorf(ylop - 1.0f));
      int h1 = min(hhi, (int)ceilf(yhip + 1.0f));
      for (int w = lane; w < WS4; w += 32) {
        float acc = 0.0f;
        if (w < WS) {
          const float* col = Fc + (wlo + w);
          for (int h = h0; h <= h1; ++h)
            acc = fmaf(sWy[p][h - hlo], col[h * Ww], acc);
        }
        sT[p][c][w] = acc;
      }
    }
  }
  __syncthreads();

  // ---- Stage B (WMMA f32 16x16x4): out[c,q] per p = sum_w T[p][c,w] * wxT[w,q]
  // A 16x4 layout: VGPR0 lane L -> A[M=L&15][K=2*(L>>4)], VGPR1 -> K+1
  // B 4x16 layout: VGPR0 lane L -> B[K=2*(L>>4)][N=L&15],  VGPR1 -> K+1
  const int m    = lane & 15;
  const int koff = (lane >> 4) * 2;
  const int pBeg = (wave == 0) ? 0 : 4;
  const int pEnd = (wave == 0) ? 4 : POOLED;
  const int nk   = WS4 >> 2;

  for (int p = pBeg; p < pEnd; ++p) {
    v8f acc = {};
    for (int k = 0; k < nk; ++k) {
      int w0 = k * 4 + koff;
      v2f Am, Bm;
      Am.x = sT[p][m][w0];
      Am.y = sT[p][m][w0 + 1];
      Bm.x = sWxT[w0][m];
      Bm.y = sWxT[w0 + 1][m];
      acc = __builtin_amdgcn_wmma_f32_16x16x4_f32(
          /*neg_a=*/false, Am, /*neg_b=*/false, Bm,
          /*c_mod=*/(short)0, acc, /*reuse_a=*/false, /*reuse_b=*/false);
    }
    // D layout: VGPR r, lane L -> D[M = r + 8*(L>>4)][N = L&15]
    int q     = lane & 15;
    int chalf = (lane >> 4) * 8;
    if (q < POOLED) {
#pragma unroll
      for (int rr = 0; rr < 8; ++rr) {
        int c = c0 + chalf + rr;
        out[(((size_t)n * Cc + c) * POOLED + p) * POOLED + q] = acc[rr] * invA;
      }
    }
  }
}

extern "C" void kernel_launch(void* const* d_in, const int* in_sizes, int n_in,
                              void* d_out, int out_size, void* d_ws, size_t ws_size,
                              hipStream_t stream) {
  const float* F    = (const float*)d_in[0];   // [2,256,100,100] fp32
  const float* rois = (const float*)d_in[1];   // [N,5] fp32
  float*       out  = (float*)d_out;           // [N,256,7,7] fp32
  const int N = in_sizes[1] / 5;

  dim3 grid(Cc / 16, N);   // (channel tiles, rois)
  prroi_pool_wmma<<<grid, 64, 0, stream>>>(F, rois, out);
}